// CRF_43387759624791
// MI455X (gfx1250) — compile-verified
//
#include <hip/hip_runtime.h>
#include <hip/hip_bf16.h>
#include <stdint.h>

// Problem constants (match reference)
#define Bv 1024
#define Tv 512
#define Kv 64
#define START_IX 62
#define STOP_IX  63
#define NEGC (-10000.0f)
#define NB 2                      // batches per block
#define NTHREADS (NB * Kv)        // 128 threads = 4 waves (wave32)

__global__ __launch_bounds__(NTHREADS)
void viterbi_crf_kernel(const float* __restrict__ feats,
                        const float* __restrict__ transitions,
                        float* __restrict__ out)
{
    // LDS: ping-pong forward var, 4-slot emission ring, full backpointers
    __shared__ __align__(16) float        s_fv[2][NB * Kv];     // 1 KB
    __shared__ __align__(16) float        s_feat[4][NB * Kv];   // 2 KB (triple+ buffered ring)
    __shared__              unsigned char s_bptr[NB * Tv * Kv]; // 64 KB (CDNA5: 320KB/WGP)

    const int tid  = threadIdx.x;
    const int next = tid & (Kv - 1);   // state this thread owns
    const int bloc = tid >> 6;         // batch slot within block
    const int b    = blockIdx.x * NB + bloc;

    // ---- Transition row for this 'next' lives entirely in VGPRs (64 regs) ----
    float tr[Kv];
    {
        const float4* trow = (const float4*)(transitions + next * Kv);
        #pragma unroll
        for (int i = 0; i < 16; ++i) {
            float4 v = trow[i];
            tr[4*i+0] = v.x; tr[4*i+1] = v.y; tr[4*i+2] = v.z; tr[4*i+3] = v.w;
        }
    }

    // ---- init forward_var (slot 0) ----
    s_fv[0][bloc * Kv + next] = (next == START_IX) ? 0.0f : NEGC;

    const uint64_t fbase   = (uint64_t)(uintptr_t)feats;
    const unsigned lanegl  = (unsigned)b * Tv * Kv * 4u + (unsigned)next * 4u; // +t*Kv*4 per step
    const unsigned laneloc = (unsigned)(bloc * Kv + next);

    // ---- prime the async ring: issue t=0 and t=1 (gfx1250 async global->LDS) ----
    {
        unsigned l0 = (unsigned)(uintptr_t)(void*)&s_feat[0][laneloc];
        unsigned g0 = lanegl + 0u * Kv * 4u;
        asm volatile("global_load_async_to_lds_b32 %0, %1, %2"
                     :: "v"(l0), "v"(g0), "s"(fbase) : "memory");
        unsigned l1 = (unsigned)(uintptr_t)(void*)&s_feat[1][laneloc];
        unsigned g1 = lanegl + 1u * Kv * 4u;
        asm volatile("global_load_async_to_lds_b32 %0, %1, %2"
                     :: "v"(l1), "v"(g1), "s"(fbase) : "memory");
    }
    // t=0 must be resident (in-order completion => cnt<=1 retires the oldest)
    asm volatile("s_wait_asynccnt 0x1" ::: "memory");
    __syncthreads();

    // ---- Viterbi forward scan ----
    for (int t = 0; t < Tv; ++t) {
        // issue prefetch 2 steps ahead into ring slot (t+2)&3 (uniform branch)
        const bool more = (t + 2 < Tv);
        if (more) {
            unsigned l = (unsigned)(uintptr_t)(void*)&s_feat[(t + 2) & 3][laneloc];
            unsigned g = lanegl + (unsigned)(t + 2) * Kv * 4u;
            asm volatile("global_load_async_to_lds_b32 %0, %1, %2"
                         :: "v"(l), "v"(g), "s"(fbase) : "memory");
        }

        // max-plus reduction over prev states; fv read as ds_load_b128 broadcasts
        const float4* fv4 = (const float4*)&s_fv[t & 1][bloc * Kv];
        float best = -3.402823466e38f;
        int   bi   = 0;
        #pragma unroll
        for (int p4 = 0; p4 < 16; ++p4) {
            float4 f = fv4[p4];
            float s;
            s = f.x + tr[4*p4+0]; if (s > best) { best = s; bi = 4*p4+0; }
            s = f.y + tr[4*p4+1]; if (s > best) { best = s; bi = 4*p4+1; }
            s = f.z + tr[4*p4+2]; if (s > best) { best = s; bi = 4*p4+2; }
            s = f.w + tr[4*p4+3]; if (s > best) { best = s; bi = 4*p4+3; }
        }

        float nf = best + s_feat[t & 3][laneloc];                  // add emission
        s_bptr[(bloc * Tv + t) * Kv + next] = (unsigned char)bi;   // LDS-resident backptr
        s_fv[(t + 1) & 1][laneloc] = nf;                           // ping-pong write

        // ensure slot (t+1)&3 is resident before anyone reads it next iteration:
        // loads retire in order, so "<=1 outstanding" retires everything up to t+1
        // while the t+2 issue stays in flight. Tail: nothing new issued -> drain to 0.
        if (more) {
            asm volatile("s_wait_asynccnt 0x1" ::: "memory");
        } else {
            asm volatile("s_wait_asynccnt 0x0" ::: "memory");
        }
        __syncthreads();                                   // fv/backptr visible to all waves
    }

    // ---- terminal scores: fv_final + transitions[STOP, :] ----
    float term = s_fv[Tv & 1][laneloc] + transitions[STOP_IX * Kv + next];
    s_feat[0][laneloc] = term;
    __syncthreads();

    // ---- per-batch argmax + backtrack (leader thread; bptr chain stays in LDS) ----
    if (next == 0) {
        float bestv = -3.402823466e38f;
        int   bt    = 0;
        for (int k = 0; k < Kv; ++k) {
            float v = s_feat[0][bloc * Kv + k];
            if (v > bestv) { bestv = v; bt = k; }   // strict > : first-max like jnp.argmax
        }
        out[b] = bestv;                              // path_score

        int tag = bt;
        float* path = out + Bv + (size_t)b * Tv;     // best_path (emitted as floats)
        for (int t = Tv - 1; t >= 0; --t) {
            path[t] = (float)tag;
            tag = (int)s_bptr[(bloc * Tv + t) * Kv + tag];
        }
        // tag == START_IX here by construction
    }
}

extern "C" void kernel_launch(void* const* d_in, const int* in_sizes, int n_in,
                              void* d_out, int out_size, void* d_ws, size_t ws_size,
                              hipStream_t stream) {
    const float* feats       = (const float*)d_in[0];   // [B, T, K] f32
    const float* transitions = (const float*)d_in[1];   // [K, K]    f32
    float* out = (float*)d_out;                         // [B] scores ++ [B, T] path
    (void)in_sizes; (void)n_in; (void)d_ws; (void)ws_size; (void)out_size;

    dim3 grid(Bv / NB);       // 512 blocks
    dim3 block(NTHREADS);     // 128 threads (4 × wave32)
    viterbi_crf_kernel<<<grid, block, 0, stream>>>(feats, transitions, out);
}